// SequenceEmbedder_5643587026959
// MI455X (gfx1250) — compile-verified
//
#include <hip/hip_runtime.h>

// ---------------------------------------------------------------------------
// SequenceEmbedder for MI455X (gfx1250, wave32).
// out[b, pos, d] = s * W[d] + bias[d]  (rank-1 GEMM -> v_wmma_f32_16x16x4_f32)
// or a token row tok[id][d] (overwritten by a second tiny kernel, ~3% traffic).
// Store-bandwidth bound (~275 MB writes -> ~12 us floor @ 23.3 TB/s):
// all output stores are non-temporal, immediate-offset, unpredicated on the
// fast path (wave-uniform partial-tile check; only 64/16768 waves are partial).
// ---------------------------------------------------------------------------

typedef __attribute__((ext_vector_type(2))) float v2f;
typedef __attribute__((ext_vector_type(4))) float v4f;
typedef __attribute__((ext_vector_type(8))) float v8f;

#define NB 64
#define NC 64
#define NR 32
#define ND 256
#define NL 2016               // T length per batch
#define LTOT 2079             // T-region sequence length
#define BAYLEN 2112           // NC*(NR+1), multiple of 16
#define SEQ 4191              // BAYLEN + LTOT
#define TILES_BAY 132         // 2112/16
#define TILES_T 130           // ceil(2079/16)
#define TILES_PER_B (TILES_BAY + TILES_T)
#define TOTAL_TILES (NB * TILES_PER_B)

// Token position j (1..62) in the T region; p(0) == -1 sentinel.
__device__ __forceinline__ int p_tok(int j) {
    return j * 65 - (j * (j + 1)) / 2 - 1;
}

__global__ void __launch_bounds__(256) emb_wmma_kernel(
    const float* __restrict__ bay, const float* __restrict__ T,
    const float* __restrict__ Wc, const float* __restrict__ bc,
    const float* __restrict__ Wt, const float* __restrict__ bt,
    float* __restrict__ out)
{
    const int lane = threadIdx.x & 31;
    const int wave = threadIdx.x >> 5;
    const int tile = blockIdx.x * 8 + wave;
    if (tile >= TOTAL_TILES) return;   // wave-uniform exit

    const int  b      = tile / TILES_PER_B;
    const int  t      = tile - b * TILES_PER_B;
    const bool isBay  = (t < TILES_BAY);
    const int  i0     = isBay ? t * 16 : (t - TILES_BAY) * 16;
    const int  regOff = isBay ? 0 : BAYLEN;
    const int  regLen = isBay ? BAYLEN : LTOT;
    const float* __restrict__ W  = isBay ? Wc : Wt;
    const float* __restrict__ Bi = isBay ? bc : bt;

    const int n = lane & 15;           // N (column) index this lane owns in C/D

    // Preload this lane's W / bias columns; static indices after full unroll.
    float wv[16], bv[16];
    #pragma unroll
    for (int ch = 0; ch < 16; ++ch) {
        wv[ch] = W[ch * 16 + n];
        bv[ch] = Bi[ch * 16 + n];
    }

    // Row scalar for A's K=0 slot: lanes 0..15 own rows i0+lane (A layout:
    // VGPR0 lanes 0-15 = K=0, M=lane). Token/padding rows -> 0 (overwritten).
    float s = 0.0f;
    if (lane < 16) {
        int i = i0 + lane;
        if (isBay) {
            int c = i / 33;
            int r = i - c * 33;
            if (r < NR) s = bay[(size_t)b * (NC * NR) + c * NR + r];
        } else if (i < LTOT) {
            // largest j in [0,62] with p_tok(j) <= i  (sqrt guess + exact fixup)
            int j = (int)((129.0f - sqrtf((float)(16633 - 8 * i))) * 0.5f);
            if (j < 0) j = 0;
            if (j > 62) j = 62;
            while (j < 62 && p_tok(j + 1) <= i) ++j;
            while (j > 0 && p_tok(j) > i) --j;
            bool tokrow = (p_tok(j) == i) || (i == LTOT - 1);
            if (!tokrow) s = T[(size_t)b * NL + (i - j)];
        }
    }

    // A operand: only K=0 nonzero; K>=1 all zero makes the result independent
    // of B's K>=1 slot placement (layout-ambiguity safe).
    v2f a;
    a.x = (lane < 16) ? s : 0.0f;      // VGPR0: K=0 (lanes 0-15), K=2 (16-31) -> 0
    a.y = 0.0f;                        // VGPR1: K=1 / K=3 -> 0

    const int mAdd = (lane >> 4) << 3; // D layout: lanes 16-31 hold rows M+8
    const size_t rowBase = (size_t)b * SEQ + regOff;

    if (i0 + 16 <= regLen) {
        // ---- fast path: full tile, no predication, one base + imm offsets ----
        float* __restrict__ base = out + (rowBase + i0 + mAdd) * (size_t)ND + n;
        #pragma unroll
        for (int ch = 0; ch < 16; ++ch) {
            v2f bm;
            bm.x = (lane < 16) ? wv[ch] : 0.0f;  // B K=0 row: N=lane
            bm.y = 0.0f;
            v8f cm;
            #pragma unroll
            for (int k = 0; k < 8; ++k) cm[k] = bv[ch];
            cm = __builtin_amdgcn_wmma_f32_16x16x4_f32(
                false, a, false, bm, (short)0, cm, false, false);
            #pragma unroll
            for (int k = 0; k < 8; ++k) {
                __builtin_nontemporal_store(cm[k], base + k * ND + ch * 16);
            }
        }
    } else {
        // ---- slow path: partial tile (last T tile per batch, 64 waves) ----
        #pragma unroll
        for (int ch = 0; ch < 16; ++ch) {
            v2f bm;
            bm.x = (lane < 16) ? wv[ch] : 0.0f;
            bm.y = 0.0f;
            v8f cm;
            #pragma unroll
            for (int k = 0; k < 8; ++k) cm[k] = bv[ch];
            cm = __builtin_amdgcn_wmma_f32_16x16x4_f32(
                false, a, false, bm, (short)0, cm, false, false);
            #pragma unroll
            for (int k = 0; k < 8; ++k) {
                int i = i0 + k + mAdd;
                if (i < regLen) {
                    __builtin_nontemporal_store(
                        cm[k], out + (rowBase + i) * (size_t)ND + ch * 16 + n);
                }
            }
        }
    }
}

// Overwrite the 127 token rows per batch (8 MB total, ~3% of output traffic).
__global__ void __launch_bounds__(64) tok_fill_kernel(
    const float* __restrict__ tok, float* __restrict__ out)
{
    int row  = blockIdx.x;             // 0 .. NB*127-1
    int b    = row / 127;
    int tIdx = row - b * 127;
    int pos, id;
    if (tIdx < NC) {                   // bay region: col-stop / bay-stop tokens
        pos = tIdx * 33 + 32;
        id  = (tIdx == NC - 1) ? 1 : 0;
    } else {                           // T region: next-port tokens + T-stop
        int j = tIdx - (NC - 1);       // 1..63
        pos = BAYLEN + ((j <= 62) ? p_tok(j) : (LTOT - 1));
        id  = (j == 63) ? 3 : 2;
    }
    const v4f* src = (const v4f*)(tok + (size_t)id * ND);
    v4f v = src[threadIdx.x];
    v4f* dst = (v4f*)(out + ((size_t)b * SEQ + pos) * ND);
    __builtin_nontemporal_store(v, dst + threadIdx.x);
}

extern "C" void kernel_launch(void* const* d_in, const int* in_sizes, int n_in,
                              void* d_out, int out_size, void* d_ws, size_t ws_size,
                              hipStream_t stream) {
    (void)in_sizes; (void)n_in; (void)out_size; (void)d_ws; (void)ws_size;
    const float* bay = (const float*)d_in[0];
    const float* T   = (const float*)d_in[1];
    const float* Wc  = (const float*)d_in[2];
    const float* bc  = (const float*)d_in[3];
    const float* Wt  = (const float*)d_in[4];
    const float* bt  = (const float*)d_in[5];
    const float* tok = (const float*)d_in[6];
    float* out = (float*)d_out;

    emb_wmma_kernel<<<(TOTAL_TILES + 7) / 8, 256, 0, stream>>>(
        bay, T, Wc, bc, Wt, bt, out);
    tok_fill_kernel<<<NB * 127, 64, 0, stream>>>(tok, out);
}